// SparseClusterLayer_58514634440869
// MI455X (gfx1250) — compile-verified
//
#include <hip/hip_runtime.h>

#define N_ROWS 65536
#define C_CLS  256
#define D_DIM  1024
#define ALPHA_F 1.0f
#define GAMMA_F 0.01f

typedef __attribute__((ext_vector_type(2))) float v2f;
typedef __attribute__((ext_vector_type(8))) float v8f;

// d_ws layout (floats):
//   accum [0, C*D)              : segment-sum accumulator (C x D)
//   pcd   [C*D, C*D + C)        : per-class denominator
//   sq    [C*D + C, C*D + 2C)   : per-class squared norm

__global__ void k_zero(float* __restrict__ ws, int total) {
    int idx = blockIdx.x * blockDim.x + threadIdx.x;
    if (idx < total) ws[idx] = 0.0f;
}

__global__ void __launch_bounds__(256) k_sq(const float* __restrict__ cluster,
                                            float* __restrict__ sq) {
    const int c = blockIdx.x;
    const int t = threadIdx.x;
    const float4 v = ((const float4*)(cluster + (size_t)c * D_DIM))[t];
    float s = v.x * v.x + v.y * v.y + v.z * v.z + v.w * v.w;
    for (int off = 16; off > 0; off >>= 1) s += __shfl_down(s, off, 32);
    __shared__ float red[8];
    const int wave = t >> 5, lane = t & 31;
    if (lane == 0) red[wave] = s;
    __syncthreads();
    if (t == 0) {
        float tot = 0.0f;
        for (int w = 0; w < 8; ++w) tot += red[w];
        sq[c] = tot;
    }
}

// One wave per 16x16 output tile of G = cluster * cluster^T.
// Fused: pcd[i] += sum_j max(sq[i] + sq[j] - 2*G[i][j], 0)
__global__ void __launch_bounds__(32) k_gram(const float* __restrict__ cluster,
                                             const float* __restrict__ sq,
                                             float* __restrict__ pcd) {
    const int tile = blockIdx.x;       // 0..255 => 16x16 grid of tiles
    const int ti = tile >> 4;
    const int tj = tile & 15;
    const int lane = threadIdx.x;      // 0..31, EXEC all ones
    const int half = lane >> 4;        // 0 or 1
    const int sub  = lane & 15;

    // ISA A-layout 16x4 f32: lane holds row (lane&15), K = k0 + 2*half + {0,1}
    // B = cluster^T (4x16): lane holds col (lane&15) -> same addressing, tj base.
    const float* __restrict__ arow = cluster + (size_t)(ti * 16 + sub) * D_DIM + 2 * half;
    const float* __restrict__ brow = cluster + (size_t)(tj * 16 + sub) * D_DIM + 2 * half;

    v8f acc = {};
    for (int k = 0; k < D_DIM; k += 4) {
        v2f a = *(const v2f*)(arow + k);
        v2f b = *(const v2f*)(brow + k);
        acc = __builtin_amdgcn_wmma_f32_16x16x4_f32(
            /*neg_a=*/false, a, /*neg_b=*/false, b,
            /*c_mod=*/(short)0, acc, /*reuse_a=*/false, /*reuse_b=*/false);
    }

    // C/D layout: VGPR r -> row (r + 8*half), col sub within tile.
    for (int r = 0; r < 8; ++r) {
        const int gi = ti * 16 + r + 8 * half;
        const int gj = tj * 16 + sub;
        float v = sq[gi] + sq[gj] - 2.0f * acc[r];
        v = v > 0.0f ? v : 0.0f;
        unsafeAtomicAdd(pcd + gi, v);
    }
}

// One block per feature row: loss + weighted-diff scatter into accum.
__global__ void __launch_bounds__(256) k_main(const float* __restrict__ features,
                                              const int* __restrict__ labels,
                                              const float* __restrict__ cluster,
                                              const float* __restrict__ cw,
                                              const float* __restrict__ pcd,
                                              float* __restrict__ accum,
                                              float* __restrict__ loss) {
    const int row = blockIdx.x;
    const int t = threadIdx.x;
    const int lab = labels[row];
    const float denom = pcd[lab] + ALPHA_F;

    const float4 f = ((const float4*)(features + (size_t)row * D_DIM))[t];
    const float4 c = ((const float4*)(cluster + (size_t)lab * D_DIM))[t];
    float4 d;
    d.x = f.x - c.x; d.y = f.y - c.y; d.z = f.z - c.z; d.w = f.w - c.w;
    float part = d.x * d.x + d.y * d.y + d.z * d.z + d.w * d.w;

    for (int off = 16; off > 0; off >>= 1) part += __shfl_down(part, off, 32);
    __shared__ float red[8];
    const int wave = t >> 5, lane = t & 31;
    if (lane == 0) red[wave] = part;
    __syncthreads();
    if (t == 0) {
        float nume = 0.0f;
        for (int w = 0; w < 8; ++w) nume += red[w];
        loss[row] = nume / denom;
    }

    const float scale = cw[lab] / denom;
    float* __restrict__ ap = accum + (size_t)lab * D_DIM + t * 4;
    unsafeAtomicAdd(ap + 0, scale * d.x);
    unsafeAtomicAdd(ap + 1, scale * d.y);
    unsafeAtomicAdd(ap + 2, scale * d.z);
    unsafeAtomicAdd(ap + 3, scale * d.w);
}

__global__ void k_final(const float* __restrict__ cluster,
                        const float* __restrict__ accum,
                        float* __restrict__ out_nc) {
    int idx = blockIdx.x * blockDim.x + threadIdx.x;
    if (idx < C_CLS * D_DIM)
        out_nc[idx] = cluster[idx] - GAMMA_F * accum[idx];
}

extern "C" void kernel_launch(void* const* d_in, const int* in_sizes, int n_in,
                              void* d_out, int out_size, void* d_ws, size_t ws_size,
                              hipStream_t stream) {
    (void)in_sizes; (void)n_in; (void)out_size; (void)ws_size;
    const float* features = (const float*)d_in[0];
    const int*   labels   = (const int*)d_in[1];
    const float* cluster  = (const float*)d_in[2];
    const float* cw       = (const float*)d_in[3];
    float* out = (float*)d_out;            // [0,N): loss ; [N, N+C*D): new_cluster
    float* ws  = (float*)d_ws;
    float* accum = ws;
    float* pcd   = ws + (size_t)C_CLS * D_DIM;
    float* sq    = pcd + C_CLS;

    const int zero_total = C_CLS * D_DIM + C_CLS;   // accum + pcd
    k_zero<<<(zero_total + 255) / 256, 256, 0, stream>>>(ws, zero_total);
    k_sq<<<C_CLS, 256, 0, stream>>>(cluster, sq);
    k_gram<<<(C_CLS / 16) * (C_CLS / 16), 32, 0, stream>>>(cluster, sq, pcd);
    k_main<<<N_ROWS, 256, 0, stream>>>(features, labels, cluster, cw, pcd, accum, out);
    k_final<<<(C_CLS * D_DIM + 255) / 256, 256, 0, stream>>>(cluster, accum, out + N_ROWS);
}